// TaylorExp_52097953300652
// MI455X (gfx1250) — compile-verified
//
#include <hip/hip_runtime.h>
#include <hip/hip_bf16.h>

typedef __attribute__((ext_vector_type(2))) float v2f;
typedef __attribute__((ext_vector_type(8))) float v8f;

#define NTOK   (8 * 16 * 4096)   // B*H*S = 524288 tokens
#define OUTW   273               // 1 + 16 + 256
#define BLOCKS 8192
#define THREADS 256
#define TPW    8                 // 65536 wave32s -> exactly 8 contiguous tokens/wave

__global__ __launch_bounds__(THREADS)
void TaylorExp_52097953300652_kernel(const float* __restrict__ x,
                                     float* __restrict__ out) {
    const int  lane = threadIdx.x & 31;
    const long wave = (long)((blockIdx.x * (int)blockDim.x + threadIdx.x) >> 5);
    const long t0   = wave * TPW;

    const float s2  = 0.17677669529663688f; // 1/(4*sqrt(2))
    const int  col  = lane & 15;
    const int  half = lane >> 4;            // 0: D rows 0..7, 1: rows 8..15
    const int  shl  = (lane + 15) & 15;     // (lane-1) mod 16 for the header shift

    const float* xbase = x + t0 * 16;

    // Stage 1: batch ALL loads up front, unconditionally (lanes 16..31 re-read
    // the same 64B row -> always in bounds, no exec branches, loads clause up).
    float xl[TPW];   // x[col]        of token i
    float hb[TPW];   // x[(lane-1)%16] of token i (header source)
#pragma unroll
    for (int i = 0; i < TPW; ++i) xl[i] = xbase[i * 16 + col];
#pragma unroll
    for (int i = 0; i < TPW; ++i) hb[i] = xbase[i * 16 + shl];

    float* obase = out + t0 * (long)OUTW;

    // Stage 2: per token: rank-1 f32 WMMA outer product + NT streaming stores.
#pragma unroll
    for (int i = 0; i < TPW; ++i) {
        // Register select (v_cndmask), not a branch: lanes 16..31 -> 0 so only
        // the K=0 column of A and row K=0 of B are nonzero.
        float xv = (lane < 16) ? xl[i] : 0.0f;

        v2f a; a.x = xv * s2; a.y = 0.0f;   // A 16x4: K=0 column = x*s2
        v2f b; b.x = xv;      b.y = 0.0f;   // B 4x16: row K=0 = x

        v8f c = {};
        c = __builtin_amdgcn_wmma_f32_16x16x4_f32(
                false, a, false, b, (short)0, c, false, false);

        float* o = obase + i * (long)OUTW;

        // Header o[0..16] = [1, x/2]: full 32-lane store, no exec mask.
        // Lanes 17..31 write scratch into o[17..31]; the x2 stores below are
        // later in program order from the same wave (stores are kept in order
        // with stores), so they overwrite those slots with correct data.
        float hv = (lane == 0) ? 1.0f : 0.5f * hb[i];
        __builtin_nontemporal_store(hv, o + lane);

        // x2 block: D vgpr k, lane l -> flat (k + 8*half)*16 + col, i.e. each
        // store covers two contiguous 64B runs; 8 stores cover o[17..272].
        float* p = o + 17 + half * 128 + col;
#pragma unroll
        for (int k = 0; k < 8; ++k)
            __builtin_nontemporal_store(c[k], p + k * 16);
    }
}

extern "C" void kernel_launch(void* const* d_in, const int* in_sizes, int n_in,
                              void* d_out, int out_size, void* d_ws, size_t ws_size,
                              hipStream_t stream) {
    const float* x   = (const float*)d_in[0];
    float*       out = (float*)d_out;
    TaylorExp_52097953300652_kernel<<<BLOCKS, THREADS, 0, stream>>>(x, out);
}